// Manual_LSTM_36842229465118
// MI455X (gfx1250) — compile-verified
//
#include <hip/hip_runtime.h>
#include <math.h>

typedef float v2f __attribute__((ext_vector_type(2)));
typedef float v4f __attribute__((ext_vector_type(4)));
typedef float v8f __attribute__((ext_vector_type(8)));

#define B_ 32
#define S_ 2048
#define I_ 256
#define H_ 256
#define K_ 512            // H + I (h rows first, then x rows, matching concat([h,x]))
#define NBLK 16           // one block per 16-column slice of H
#define NT 16             // H-columns owned per block
#define THREADS 256       // 8 wave32

// LDS strides in floats, chosen so WMMA b64 fragment loads are bank-conflict-free
#define WROW 516          // 512 + 4  (stride%64 == 4 -> banks 4*row + k, all distinct)
#define AROW 516
#define HXBUF (32 * AROW) // one [h|x] staging buffer (double-buffered)

#define OFF_W   0
#define OFF_HX  (64 * WROW)             // 64 transposed weight rows (4 gates x 16 cols)
#define OFF_G   (OFF_HX + 2 * HXBUF)    // 2 x staged [h|x] tiles (x double-buffered)
#define OFF_C   (OFF_G + 8 * 256)       // 8 waves x 16x16 gate tiles
#define LDS_FLOATS (OFF_C + 32 * 16)    // + 32x16 persistent cell-state slab (~274 KB)

extern "C" __global__ void __launch_bounds__(THREADS, 1)
lstm_persistent(const float* __restrict__ x,
                const float* __restrict__ h0,
                const float* __restrict__ c0,
                const float* __restrict__ Wf, const float* __restrict__ bf,
                const float* __restrict__ Wi, const float* __restrict__ bi,
                const float* __restrict__ Wc, const float* __restrict__ bc,
                const float* __restrict__ Wo, const float* __restrict__ bo,
                float* __restrict__ out,
                unsigned int* __restrict__ barrier_cnt)
{
    extern __shared__ float lds[];
    float* sWt = lds + OFF_W;   // [64][WROW]  sWt[g*16+nl][k] = W_g[k][n0+nl]
    float* sHX = lds + OFF_HX;  // 2x [32][AROW]: [b][0..255]=h_prev, [b][256..511]=x_t
    float* sG  = lds + OFF_G;   // [8][16][16] post-activation gate tiles per wave
    float* sC  = lds + OFF_C;   // [32][16]    cell state for this block's column slice

    const int tid = threadIdx.x;
    const int blk = blockIdx.x;
    const int n0  = blk * NT;

    // ---------- one-time: transpose-load this block's weight slab into LDS ----------
    for (int idx = tid; idx < 64 * K_; idx += THREADS) {
        int nl = idx & 63;
        int k  = idx >> 6;
        int g  = nl >> 4;
        const float* wp = (g == 0) ? Wf : (g == 1) ? Wi : (g == 2) ? Wc : Wo;
        sWt[nl * WROW + k] = wp[(size_t)k * H_ + n0 + (nl & 15)];
    }
    // ---------- one-time: init cell state slab from c0 ----------
    for (int e = tid; e < B_ * NT; e += THREADS) {
        int b = e >> 4, nl = e & 15;
        sC[e] = c0[b * H_ + n0 + nl];
    }
    // ---------- one-time: stage x(0) into buffer 0 (vectorized b128) ----------
    for (int idx = tid; idx < B_ * (I_ / 4); idx += THREADS) {
        int b = idx >> 6, j = idx & 63;                   // j: float4 index within row
        v4f v = *(const v4f*)(x + ((size_t)b * S_) * I_ + 4 * j);
        *(v4f*)(sHX + b * AROW + H_ + 4 * j) = v;
    }

    // wave -> tile mapping
    const int lane = tid & 31;
    const int wave = tid >> 5;        // 0..7
    const int mt   = wave >> 2;       // row-tile: b rows [mt*16, mt*16+16)
    const int nt   = wave & 3;        // gate id: 0=f 1=i 2=cand 3=o
    const int i16  = lane & 15;       // M (A) / N (B) index
    const int kp   = lane >> 4;       // K-pair selector
    const int aoff = (mt * 16 + i16) * AROW + 2 * kp;
    // hoisted per-buffer A-fragment base pointers (loop-invariant bases -> DS imm offsets)
    const float* aptr0 = sHX + aoff;
    const float* aptr1 = sHX + HXBUF + aoff;
    const float* bptr  = sWt + (nt * 16 + i16) * WROW + 2 * kp;
    const float* bgp   = (nt == 0) ? bf : (nt == 1) ? bi : (nt == 2) ? bc : bo;
    const float  bias  = bgp[n0 + i16];

    // wave-uniform activation constants: gates 0,1,3 -> sigmoid; gate 2 -> tanh
    const float act_kin  = (nt == 2) ? -2.0f : -1.0f;
    const float act_mulo = (nt == 2) ?  2.0f :  1.0f;
    const float act_addo = (nt == 2) ? -1.0f :  0.0f;

    for (int t = 0; t < S_; ++t) {
        float* hx_cur = sHX + (t & 1) * HXBUF;
        float* hx_nxt = sHX + ((t + 1) & 1) * HXBUF;

        // ---------- critical path: stage h(t-1) (device-scope, cross-WGP coherent) ----------
        for (int idx = tid; idx < B_ * H_; idx += THREADS) {
            int b = idx >> 8;
            int k = idx & 255;
            const float* hp = (t == 0)
                ? (h0 + b * H_ + k)
                : (out + ((size_t)b * S_ + (t - 1)) * H_ + k);
            hx_cur[b * AROW + k] =
                __hip_atomic_load(hp, __ATOMIC_RELAXED, __HIP_MEMORY_SCOPE_AGENT);
        }
        __syncthreads();

        // ---------- g[mt-rows, gate nt] = [h|x] @ Wslab : 128 x V_WMMA_F32_16X16X4_F32 ----------
        const float* aptr = (t & 1) ? aptr1 : aptr0;
        v8f acc = {};
        #pragma unroll 8
        for (int k = 0; k < K_; k += 4) {
            v2f a = *(const v2f*)(aptr + k);
            v2f b = *(const v2f*)(bptr + k);
            acc = __builtin_amdgcn_wmma_f32_16x16x4_f32(
                      false, a, false, b, (short)0, acc, false, false);
        }

        // ---------- bias + branchless activation, tile to LDS (M = v + 8*kp, N = i16) ----------
        float* gdst = sG + wave * 256 + kp * 128 + i16;
        #pragma unroll
        for (int v = 0; v < 8; ++v) {
            float g = acc[v] + bias;
            float s = __builtin_amdgcn_rcpf(1.0f + __expf(act_kin * g));
            gdst[v * 16] = __builtin_fmaf(act_mulo, s, act_addo);
        }
        __syncthreads();

        // ---------- elementwise cell/hidden update for this block's 32x16 slice ----------
        for (int e = tid; e < B_ * NT; e += THREADS) {
            int b = e >> 4, nl = e & 15;
            int mtt = b >> 4, m = b & 15;
            const float* gg = sG + mtt * 4 * 256 + m * 16 + nl;
            float f    = gg[0];
            float ig   = gg[256];
            float cand = gg[512];
            float o    = gg[768];
            float cn = __builtin_fmaf(sC[e], f, ig * cand);
            sC[e] = cn;
            float th = __builtin_fmaf(2.0f,
                          __builtin_amdgcn_rcpf(1.0f + __expf(-2.0f * cn)), -1.0f);
            float hn = o * th;
            __hip_atomic_store(out + ((size_t)b * S_ + t) * H_ + n0 + nl, hn,
                               __ATOMIC_RELAXED, __HIP_MEMORY_SCOPE_AGENT);
            if (t == S_ - 1) {
                out[(size_t)B_ * S_ * H_ + (size_t)b * H_ + n0 + nl] = hn;            // h_f
                out[(size_t)B_ * S_ * H_ + (size_t)B_ * H_ + (size_t)b * H_ + n0 + nl] = cn; // c_f
            }
        }

        // ---------- off critical path: prefetch x(t+1) into the other buffer ----------
        if (t + 1 < S_) {
            for (int idx = tid; idx < B_ * (I_ / 4); idx += THREADS) {
                int b = idx >> 6, j = idx & 63;
                v4f v = *(const v4f*)(x + ((size_t)b * S_ + (t + 1)) * I_ + 4 * j);
                *(v4f*)(hx_nxt + b * AROW + H_ + 4 * j) = v;
            }
        }

        // ---------- grid-wide barrier (all 16 blocks resident: 1 WG/WGP) ----------
        __threadfence();
        __syncthreads();
        if (tid == 0) {
            atomicAdd(barrier_cnt, 1u);
            const unsigned int target = (unsigned int)(NBLK * (t + 1));
            while (__hip_atomic_load(barrier_cnt, __ATOMIC_RELAXED,
                                     __HIP_MEMORY_SCOPE_AGENT) < target) {
                __builtin_amdgcn_s_sleep(1);
            }
        }
        __syncthreads();
    }
}

extern "C" void kernel_launch(void* const* d_in, const int* in_sizes, int n_in,
                              void* d_out, int out_size, void* d_ws, size_t ws_size,
                              hipStream_t stream) {
    (void)in_sizes; (void)n_in; (void)out_size; (void)ws_size;
    const float* x  = (const float*)d_in[0];
    const float* h0 = (const float*)d_in[1];
    const float* c0 = (const float*)d_in[2];
    const float* Wf = (const float*)d_in[3];
    const float* bf = (const float*)d_in[4];
    const float* Wi = (const float*)d_in[5];
    const float* bi = (const float*)d_in[6];
    const float* Wc = (const float*)d_in[7];
    const float* bc = (const float*)d_in[8];
    const float* Wo = (const float*)d_in[9];
    const float* bo = (const float*)d_in[10];
    float* out = (float*)d_out;
    unsigned int* cnt = (unsigned int*)d_ws;

    // reset global barrier counter each call (graph-capturable memset node)
    hipMemsetAsync(d_ws, 0, sizeof(unsigned int), stream);

    const size_t lds_bytes = (size_t)LDS_FLOATS * sizeof(float);  // ~274 KB (<320 KB/WGP)
    hipFuncSetAttribute((const void*)lstm_persistent,
                        hipFuncAttributeMaxDynamicSharedMemorySize, (int)lds_bytes);

    lstm_persistent<<<NBLK, THREADS, lds_bytes, stream>>>(
        x, h0, c0, Wf, bf, Wi, bi, Wc, bc, Wo, bo, out, cnt);
}